// SampleBlock_53085795779114
// MI455X (gfx1250) — compile-verified
//
#include <hip/hip_runtime.h>
#include <hip/hip_bf16.h>
#include <math.h>

typedef float v2f __attribute__((ext_vector_type(2)));
typedef float v8f __attribute__((ext_vector_type(8)));

#define HW 4096    // 64*64
#define PTOT 16384 // 4*64*64

// ---------------- elementwise: feat_cat[:, :256] = relu(cur_feat) ----------------
__global__ void relu_copy_kernel(const float* __restrict__ src, float* __restrict__ dst) {
    int i = blockIdx.x * blockDim.x + threadIdx.x;   // over N*256*HW
    int hw = i & (HW - 1);
    int rest = i >> 12;          // img*256 + c
    int c = rest & 255;
    int img = rest >> 8;
    float v = src[i];
    dst[((img * 512 + c) << 12) + hw] = fmaxf(v, 0.f);
}

// ---------------- implicit-GEMM 3x3 conv with V_WMMA_F32_16X16X4_F32 ----------------
// Each wave computes MT M-tiles (16 rows) x 2 N-tiles (16 cols). BM=64*MT, BN=64, BK=16.
// LDS is K-pair-major (v2f) so every WMMA fragment is one aligned ds_load_b64.
// Double-buffered: one barrier per K-chunk; next chunk staged while WMMAs run.
template <int MT>
__launch_bounds__(256)
__global__ void conv3x3_wmma_kernel(const float* __restrict__ x,
                                    const float* __restrict__ wgt,
                                    const float* __restrict__ bias,
                                    float* __restrict__ y,
                                    int CI, int CO, int co_stride, int co_off,
                                    int dil, int do_relu) {
    constexpr int AP = 64 * MT + 16;   // A pair-row pitch in v2f (stride % 64 banks == 32)
    constexpr int BP = 64 + 16;        // B pair-row pitch in v2f
    __shared__ v2f As[2][8][AP];
    __shared__ v2f Bs[2][8][BP];

    const int tid  = threadIdx.x;
    const int lane = tid & 31;
    const int wv   = tid >> 5;     // 8 waves
    const int wm   = wv & 3;       // M quarter -> tiles wm*MT .. wm*MT+MT-1
    const int wn   = wv >> 2;      // N half (32 cols)
    const int lhi  = lane >> 4;    // selects K pair within a 4-K step
    const int ml   = lane & 15;
    const int p_base  = blockIdx.x * 64;
    const int co_base = blockIdx.y * 64 * MT;
    const int Ktot = CI * 9;       // multiple of 16 for CI in {256,512}

    v8f acc[MT][2];
    #pragma unroll
    for (int t = 0; t < MT; ++t) { acc[t][0] = (v8f){}; acc[t][1] = (v8f){}; }

    // ---- hoisted per-thread staging invariants ----
    int aw_base[4 * MT], aw_off[4 * MT];
    #pragma unroll
    for (int i = 0; i < 4 * MT; ++i) {
        int e  = tid + i * 256;                 // e = kk*(64*MT) + m
        int kk = e / (64 * MT);
        int m  = e - kk * (64 * MT);
        int co = co_base + m;
        aw_base[i] = (co < CO) ? co * Ktot + kk : -1;
        aw_off[i]  = (kk >> 1) * (AP * 2) + m * 2 + (kk & 1);   // float index in As[buf]
    }
    int b_h[4], b_w[4], b_img[4], b_off[4];
    #pragma unroll
    for (int i = 0; i < 4; ++i) {
        int e  = tid + i * 256;                 // e = kk*64 + n
        int kk = e >> 6;
        int n  = e & 63;
        int p  = p_base + n;
        b_img[i] = p >> 12;
        int hw   = p & (HW - 1);
        b_h[i] = hw >> 6;
        b_w[i] = hw & 63;
        b_off[i] = (kk >> 1) * (BP * 2) + n * 2 + (kk & 1);     // float index in Bs[buf]
    }

    auto stageA = [&](int k0, int buf) {
        float* af = (float*)&As[buf][0][0];
        #pragma unroll
        for (int i = 0; i < 4 * MT; ++i) {
            float v = (aw_base[i] >= 0) ? wgt[aw_base[i] + k0] : 0.f;
            af[aw_off[i]] = v;
        }
    };
    auto stageB = [&](int k0, int buf) {
        float* bf = (float*)&Bs[buf][0][0];
        #pragma unroll
        for (int i = 0; i < 4; ++i) {
            int e  = tid + i * 256;
            int kk = e >> 6;
            int k  = k0 + kk;
            int ci = k / 9;
            int r  = k - ci * 9;
            int kh = r / 3;
            int kw = r - kh * 3;
            int ih = b_h[i] + (kh - 1) * dil;
            int iw = b_w[i] + (kw - 1) * dil;
            float v = 0.f;
            if ((unsigned)ih < 64u && (unsigned)iw < 64u)
                v = x[(b_img[i] * CI + ci) * HW + (ih << 6) + iw];
            bf[b_off[i]] = v;
        }
    };

    stageA(0, 0);
    stageB(0, 0);
    __syncthreads();

    int pb = 0;
    for (int k0 = 0; k0 < Ktot; k0 += 16, pb ^= 1) {
        if (k0 + 16 < Ktot) {           // stage next chunk while computing this one
            stageA(k0 + 16, pb ^ 1);
            stageB(k0 + 16, pb ^ 1);
        }
        #pragma unroll
        for (int ks = 0; ks < 16; ks += 4) {
            int row = (ks >> 1) + lhi;  // K pair row for this lane half
            v2f b0 = Bs[pb][row][wn * 32 + ml];
            v2f b1 = Bs[pb][row][wn * 32 + 16 + ml];
            #pragma unroll
            for (int t = 0; t < MT; ++t) {
                v2f a = As[pb][row][(wm * MT + t) * 16 + ml];
                acc[t][0] = __builtin_amdgcn_wmma_f32_16x16x4_f32(false, a, false, b0,
                                                                  (short)0, acc[t][0], false, false);
                acc[t][1] = __builtin_amdgcn_wmma_f32_16x16x4_f32(false, a, false, b1,
                                                                  (short)0, acc[t][1], false, false);
            }
        }
        __syncthreads();
    }

    // ---- store D (C/D layout: vgpr r -> M = r + 8*(lane>=16), lane&15 -> N) ----
    const int mhalf = lhi * 8;
    #pragma unroll
    for (int t = 0; t < MT; ++t) {
        #pragma unroll
        for (int tt = 0; tt < 2; ++tt) {
            int p   = p_base + wn * 32 + tt * 16 + ml;
            int img = p >> 12;
            int hw  = p & (HW - 1);
            #pragma unroll
            for (int r = 0; r < 8; ++r) {
                int co = co_base + (wm * MT + t) * 16 + mhalf + r;
                if (co < CO) {
                    float v = acc[t][tt][r] + (bias ? bias[co] : 0.f);
                    if (do_relu) v = fmaxf(v, 0.f);
                    y[((img * co_stride + co_off + co) << 12) + hw] = v;
                }
            }
        }
    }
}

// ---------------- deformable conv as GEMM with on-the-fly bilinear B-tile ----------------
// M = 256 in one block (MT=4) so the expensive bilinear B-tile is sampled once per tile.
// sup: (N,256,64,64), wdef: (256,256,3,3)->(o,2304), ow: (N,73,64,64), out: (N,256,64,64)
__launch_bounds__(256)
__global__ void deform_gemm_kernel(const float* __restrict__ sup,
                                   const float* __restrict__ wdef,
                                   const float* __restrict__ ow,
                                   float* __restrict__ out) {
    constexpr int MT = 4;
    constexpr int AP = 64 * MT + 16;
    constexpr int BP = 64 + 16;
    __shared__ v2f As[2][8][AP];
    __shared__ v2f Bs[2][8][BP];

    const int tid  = threadIdx.x;
    const int lane = tid & 31;
    const int wv   = tid >> 5;
    const int wm   = wv & 3;
    const int wn   = wv >> 2;
    const int lhi  = lane >> 4;
    const int ml   = lane & 15;
    const int p_base = blockIdx.x * 64;
    const int Ktot = 2304;   // 256 channels * 9 taps

    v8f acc[MT][2];
    #pragma unroll
    for (int t = 0; t < MT; ++t) { acc[t][0] = (v8f){}; acc[t][1] = (v8f){}; }

    int aw_base[4 * MT], aw_off[4 * MT];
    #pragma unroll
    for (int i = 0; i < 4 * MT; ++i) {
        int e  = tid + i * 256;
        int kk = e / (64 * MT);
        int m  = e - kk * (64 * MT);
        aw_base[i] = m * Ktot + kk;             // o = m (BM == CO == 256)
        aw_off[i]  = (kk >> 1) * (AP * 2) + m * 2 + (kk & 1);
    }
    int b_h[4], b_w[4], b_supbase[4], b_owbase[4], b_off[4];
    #pragma unroll
    for (int i = 0; i < 4; ++i) {
        int e  = tid + i * 256;
        int kk = e >> 6;
        int n  = e & 63;
        int p  = p_base + n;
        int img = p >> 12;
        int hw  = p & (HW - 1);
        b_h[i] = hw >> 6;
        b_w[i] = hw & 63;
        b_supbase[i] = (img * 256) << 12;       // + c*HW later
        b_owbase[i]  = ((img * 73) << 12) + hw; // + ch*HW later
        b_off[i] = (kk >> 1) * (BP * 2) + n * 2 + (kk & 1);
    }

    auto stageA = [&](int k0, int buf) {
        float* af = (float*)&As[buf][0][0];
        #pragma unroll
        for (int i = 0; i < 4 * MT; ++i)
            af[aw_off[i]] = wdef[aw_base[i] + k0];
    };
    auto stageB = [&](int k0, int buf) {
        float* bf = (float*)&Bs[buf][0][0];
        #pragma unroll
        for (int i = 0; i < 4; ++i) {
            int e  = tid + i * 256;
            int kk = e >> 6;
            int k  = k0 + kk;
            int c  = k / 9;
            int kt = k - c * 9;          // tap 0..8
            int g  = c >> 6;             // deform group
            int och = (g * 9 + kt) * 2;  // offset channels in the 73-ch pred output
            float offy = ow[b_owbase[i] + (och << 12)];
            float offx = ow[b_owbase[i] + ((och + 1) << 12)];
            int ky = kt / 3;
            int kx = kt - ky * 3;
            float ys = (float)(b_h[i] - 1 + ky) + offy;
            float xs = (float)(b_w[i] - 1 + kx) + offx;
            float y0f = floorf(ys), x0f = floorf(xs);
            float wy = ys - y0f, wx = xs - x0f;
            int y0 = (int)y0f, x0 = (int)x0f;
            const float* base = sup + b_supbase[i] + (c << 12);
            float v = 0.f;
            #pragma unroll
            for (int dy = 0; dy < 2; ++dy) {
                #pragma unroll
                for (int dx = 0; dx < 2; ++dx) {
                    int yi = y0 + dy, xi = x0 + dx;
                    bool ok = ((unsigned)yi < 64u) && ((unsigned)xi < 64u);
                    int yc = min(max(yi, 0), 63);
                    int xc = min(max(xi, 0), 63);
                    float fv = base[(yc << 6) + xc];
                    float wt = (dy ? wy : 1.f - wy) * (dx ? wx : 1.f - wx);
                    v += ok ? fv * wt : 0.f;
                }
            }
            bf[b_off[i]] = v;
        }
    };

    stageA(0, 0);
    stageB(0, 0);
    __syncthreads();

    int pb = 0;
    for (int k0 = 0; k0 < Ktot; k0 += 16, pb ^= 1) {
        if (k0 + 16 < Ktot) {
            stageA(k0 + 16, pb ^ 1);
            stageB(k0 + 16, pb ^ 1);
        }
        #pragma unroll
        for (int ks = 0; ks < 16; ks += 4) {
            int row = (ks >> 1) + lhi;
            v2f b0 = Bs[pb][row][wn * 32 + ml];
            v2f b1 = Bs[pb][row][wn * 32 + 16 + ml];
            #pragma unroll
            for (int t = 0; t < MT; ++t) {
                v2f a = As[pb][row][(wm * MT + t) * 16 + ml];
                acc[t][0] = __builtin_amdgcn_wmma_f32_16x16x4_f32(false, a, false, b0,
                                                                  (short)0, acc[t][0], false, false);
                acc[t][1] = __builtin_amdgcn_wmma_f32_16x16x4_f32(false, a, false, b1,
                                                                  (short)0, acc[t][1], false, false);
            }
        }
        __syncthreads();
    }

    const int mhalf = lhi * 8;
    #pragma unroll
    for (int t = 0; t < MT; ++t) {
        #pragma unroll
        for (int tt = 0; tt < 2; ++tt) {
            int p   = p_base + wn * 32 + tt * 16 + ml;
            int img = p >> 12;
            int hw  = p & (HW - 1);
            #pragma unroll
            for (int r = 0; r < 8; ++r) {
                int o = (wm * MT + t) * 16 + mhalf + r;
                out[((img * 256 + o) << 12) + hw] = acc[t][tt][r];
            }
        }
    }
}

// ---------------- relu(weight) output ----------------
__global__ void wrelu_kernel(const float* __restrict__ ow, float* __restrict__ dst) {
    int i = blockIdx.x * blockDim.x + threadIdx.x;   // N*HW
    int img = i >> 12;
    int hw  = i & (HW - 1);
    float v = ow[((img * 73 + 72) << 12) + hw];
    dst[i] = fmaxf(v, 0.f);
}

extern "C" void kernel_launch(void* const* d_in, const int* in_sizes, int n_in,
                              void* d_out, int out_size, void* d_ws, size_t ws_size,
                              hipStream_t stream) {
    const float* cur    = (const float*)d_in[0];
    const float* sup    = (const float*)d_in[1];
    const float* w_d1   = (const float*)d_in[2];
    const float* b_d1   = (const float*)d_in[3];
    const float* w_d2   = (const float*)d_in[4];
    const float* b_d2   = (const float*)d_in[5];
    const float* w_d3   = (const float*)d_in[6];
    const float* b_d3   = (const float*)d_in[7];
    const float* w_lat  = (const float*)d_in[8];
    const float* b_lat  = (const float*)d_in[9];
    const float* w_pred = (const float*)d_in[10];
    const float* b_pred = (const float*)d_in[11];
    const float* w_def  = (const float*)d_in[12];

    float* feat_cat = (float*)d_ws;                         // 4*512*4096 floats
    float* lat      = feat_cat + (size_t)4 * 512 * HW;      // 4*256*4096 floats
    float* ow       = lat      + (size_t)4 * 256 * HW;      // 4*73*4096 floats

    float* out_sampled = (float*)d_out;                     // (4,256,64,64)
    float* out_w       = out_sampled + (size_t)4 * 256 * HW;// (4,1,64,64)

    dim3 blk(256);

    // feat_cat[:, :256] = relu(cur_feat)
    relu_copy_kernel<<<(4 * 256 * HW) / 256, blk, 0, stream>>>(cur, feat_cat);
    // d1/d2/d3 -> feat_cat channels 256.. with relu (single M-block each)
    conv3x3_wmma_kernel<2><<<dim3(PTOT / 64, 1), blk, 0, stream>>>(sup, w_d1, b_d1, feat_cat, 256, 128, 512, 256, 1, 1);
    conv3x3_wmma_kernel<1><<<dim3(PTOT / 64, 1), blk, 0, stream>>>(sup, w_d2, b_d2, feat_cat, 256,  64, 512, 384, 2, 1);
    conv3x3_wmma_kernel<1><<<dim3(PTOT / 64, 1), blk, 0, stream>>>(sup, w_d3, b_d3, feat_cat, 256,  64, 512, 448, 3, 1);
    // lat = relu(conv(feat_cat)) : BM=256 in one block -> B tile staged once
    conv3x3_wmma_kernel<4><<<dim3(PTOT / 64, 1), blk, 0, stream>>>(feat_cat, w_lat, b_lat, lat, 512, 256, 256, 0, 1, 1);
    // ow = conv(lat), 73 channels (BM=128 with guards), no relu
    conv3x3_wmma_kernel<2><<<dim3(PTOT / 64, 1), blk, 0, stream>>>(lat, w_pred, b_pred, ow, 256, 73, 73, 0, 1, 0);
    // deformable sampling + einsum GEMM (bilinear sampled once per spatial tile)
    deform_gemm_kernel<<<dim3(PTOT / 64, 1), blk, 0, stream>>>(sup, w_def, ow, out_sampled);
    // relu(weight)
    wrelu_kernel<<<PTOT / 256, blk, 0, stream>>>(ow, out_w);
}